// InstanceSegmentationLoss_55843164782816
// MI455X (gfx1250) — compile-verified
//
#include <hip/hip_runtime.h>
#include <hip/hip_bf16.h>

typedef __attribute__((ext_vector_type(2))) float v2f;
typedef __attribute__((ext_vector_type(8))) float v8f;

#define KIDS 64
#define NBINS (KIDS * KIDS)

// ---------------------------------------------------------------------------
// Kernel 0: zero the global joint histogram (deterministic per call).
// ---------------------------------------------------------------------------
__global__ __launch_bounds__(256) void zero_hist_kernel(unsigned* __restrict__ ghist) {
    int i = blockIdx.x * blockDim.x + threadIdx.x;
    if (i < NBINS) ghist[i] = 0u;
}

// ---------------------------------------------------------------------------
// Kernel 1: joint 64x64 histogram of (pred_id, true_id).
// Memory-bound: 8 MB of int32 reads, one ds_add_u32 per pixel into a 16KB
// per-block LDS histogram, sparse global_atomic_add flush.
// ---------------------------------------------------------------------------
__global__ __launch_bounds__(256) void joint_hist_kernel(const int* __restrict__ pred,
                                                         const int* __restrict__ tru,
                                                         unsigned* __restrict__ ghist,
                                                         int npix) {
    __shared__ unsigned lhist[NBINS];
    for (int i = threadIdx.x; i < NBINS; i += blockDim.x) lhist[i] = 0u;
    __syncthreads();

    const int gid    = blockIdx.x * blockDim.x + threadIdx.x;
    const int stride = gridDim.x * blockDim.x;
    const int nchunk = npix >> 2;

    const int4* p4 = (const int4*)pred;
    const int4* t4 = (const int4*)tru;
    for (int c = gid; c < nchunk; c += stride) {
        int4 p = p4[c];   // global_load_b128
        int4 t = t4[c];   // global_load_b128
        atomicAdd(&lhist[(p.x << 6) + t.x], 1u);
        atomicAdd(&lhist[(p.y << 6) + t.y], 1u);
        atomicAdd(&lhist[(p.z << 6) + t.z], 1u);
        atomicAdd(&lhist[(p.w << 6) + t.w], 1u);
    }
    // tail (npix not multiple of 4)
    for (int i = (nchunk << 2) + gid; i < npix; i += stride) {
        atomicAdd(&lhist[(pred[i] << 6) + tru[i]], 1u);
    }
    __syncthreads();

    for (int b = threadIdx.x; b < NBINS; b += blockDim.x) {
        unsigned v = lhist[b];
        if (v) atomicAdd(&ghist[b], v);  // non-returning global_atomic_add_u32
    }
}

// ---------------------------------------------------------------------------
// Kernel 2: finalize. One block, 64 threads (2 waves).
//   wave 0: area_p = H * ones      via V_WMMA_F32_16X16X4_F32 (row sums)
//   wave 1: area_t = ones^T * H    via V_WMMA_F32_16X16X4_F32 (col sums)
// Then per-id IoU row/col max and the scalar loss.
// ---------------------------------------------------------------------------
__global__ __launch_bounds__(64) void finalize_kernel(const unsigned* __restrict__ ghist,
                                                      float* __restrict__ out) {
    __shared__ float H[NBINS];     // joint histogram as f32 (exact: counts < 2^24)
    __shared__ float areaP[KIDS];
    __shared__ float areaT[KIDS];
    __shared__ float lsum[KIDS];
    __shared__ int   csum[KIDS];

    const int tid = threadIdx.x;
    for (int i = tid; i < NBINS; i += 64) H[i] = (float)ghist[i];
    __syncthreads();

    const int lane = tid & 31;
    const int m    = lane & 15;   // row (A) / col (B) index within the tile
    const int half = lane >> 4;   // selects K pair per ISA A/B f32 layout

    v2f ones; ones.x = 1.0f; ones.y = 1.0f;  // layout-invariant operand

    if (tid < 32) {
        // wave 0: row sums -> area_p.  A = H tile (16x4), B = ones(4x16).
        for (int rb = 0; rb < 4; ++rb) {
            v8f acc = {};
            for (int kc = 0; kc < 16; ++kc) {
                v2f a;
                a.x = H[(rb * 16 + m) * 64 + kc * 4 + 2 * half + 0];
                a.y = H[(rb * 16 + m) * 64 + kc * 4 + 2 * half + 1];
                acc = __builtin_amdgcn_wmma_f32_16x16x4_f32(
                    false, a, false, ones, (short)0, acc, false, false);
            }
            // D[m][n] == area_p[rb*16+m] for all n.
            // lane 0 VGPR r -> M=r ; lane 16 VGPR r -> M=8+r
            if (lane == 0)  { for (int r = 0; r < 8; ++r) areaP[rb * 16 + r]     = acc[r]; }
            if (lane == 16) { for (int r = 0; r < 8; ++r) areaP[rb * 16 + 8 + r] = acc[r]; }
        }
    } else {
        // wave 1: col sums -> area_t.  A = ones(16x4), B = H tile (4x16).
        for (int cb = 0; cb < 4; ++cb) {
            v8f acc = {};
            for (int kc = 0; kc < 16; ++kc) {
                v2f b;
                b.x = H[(kc * 4 + 2 * half + 0) * 64 + cb * 16 + m];
                b.y = H[(kc * 4 + 2 * half + 1) * 64 + cb * 16 + m];
                acc = __builtin_amdgcn_wmma_f32_16x16x4_f32(
                    false, ones, false, b, (short)0, acc, false, false);
            }
            // D[m][n] == area_t[cb*16+n] for all m; lane L<16 holds N=L in every VGPR.
            if (lane < 16) areaT[cb * 16 + lane] = acc[0];
        }
    }
    __syncthreads();

    // Per-id best-match IoU. Instance id 0 is background (skipped).
    float loss = 0.0f;
    int   cnt  = 0;
    if (tid >= 1) {
        float ap = areaP[tid];
        if (ap > 0.0f) {
            float mx = 0.0f;
            for (int j = 1; j < 64; ++j) {
                float in = H[tid * 64 + j];
                float u  = ap + areaT[j] - in;
                float io = (u > 0.0f) ? (in / u) : 0.0f;
                mx = fmaxf(mx, io);
            }
            loss += 1.0f - mx;
            ++cnt;
        }
        float at = areaT[tid];
        if (at > 0.0f) {
            float mx = 0.0f;
            for (int r = 1; r < 64; ++r) {
                float in = H[r * 64 + tid];
                float u  = areaP[r] + at - in;
                float io = (u > 0.0f) ? (in / u) : 0.0f;
                mx = fmaxf(mx, io);
            }
            loss += 1.0f - mx;
            ++cnt;
        }
    }
    lsum[tid] = loss;
    csum[tid] = cnt;
    __syncthreads();

    if (tid == 0) {
        float tl = 0.0f;
        int   tc = 0;
        for (int i = 0; i < 64; ++i) { tl += lsum[i]; tc += csum[i]; }
        out[0] = (tc > 0) ? (tl / (float)tc) : 0.0f;
    }
}

// ---------------------------------------------------------------------------
extern "C" void kernel_launch(void* const* d_in, const int* in_sizes, int n_in,
                              void* d_out, int out_size, void* d_ws, size_t ws_size,
                              hipStream_t stream) {
    const int* pred = (const int*)d_in[0];
    const int* tru  = (const int*)d_in[1];
    const int  npix = in_sizes[0];

    unsigned* ghist = (unsigned*)d_ws;   // 4096 * 4 = 16 KB of scratch
    float*    out   = (float*)d_out;

    zero_hist_kernel<<<dim3((NBINS + 255) / 256), dim3(256), 0, stream>>>(ghist);
    joint_hist_kernel<<<dim3(128), dim3(256), 0, stream>>>(pred, tru, ghist, npix);
    finalize_kernel<<<dim3(1), dim3(64), 0, stream>>>(ghist, out);
}